// PKRNN_2CM_36146444763937
// MI455X (gfx1250) — compile-verified
//
#include <hip/hip_runtime.h>
#include <hip/hip_bf16.h>

// Problem constants (from reference)
#define B_    256
#define T_    512
#define L_    40
#define ED_   8
#define NC_   22
#define H_    64
#define KTOT  96      // 31 (xi = 30 features + vfb) + 64 (h) + 1 zero pad
#define NTOT  256     // r(64) | z(64) | inn(64) | hn(64)
#define ROWS  16      // batch rows per workgroup (one WMMA M-tile)
#define SCAN_THREADS 128  // 4 wave32s

typedef __attribute__((ext_vector_type(2))) float v2f;
typedef __attribute__((ext_vector_type(8))) float v8f;

// ---------------------------------------------------------------------------
// Kernel 1: embedding gather-sum  esum[b,t,0:8] = sum_l emb[cat[b,t,l], :]
// Fully parallel, L2-resident table (30000x8 f32 = 960KB).
// ---------------------------------------------------------------------------
__global__ __launch_bounds__(256)
void pk_esum_kernel(const int* __restrict__ cat, const float* __restrict__ emb,
                    float* __restrict__ esum) {
    int gid = blockIdx.x * blockDim.x + threadIdx.x;   // one thread per (b,t)
    if (gid >= B_ * T_) return;
    const int* cp = cat + (size_t)gid * L_;
    float a0 = 0.f, a1 = 0.f, a2 = 0.f, a3 = 0.f;
    float a4 = 0.f, a5 = 0.f, a6 = 0.f, a7 = 0.f;
    for (int l = 0; l < L_; ++l) {
        int idx = cp[l];
        const float4* e4 = (const float4*)(emb + (size_t)idx * ED_);
        float4 e0 = e4[0];
        float4 e1 = e4[1];
        a0 += e0.x; a1 += e0.y; a2 += e0.z; a3 += e0.w;
        a4 += e1.x; a5 += e1.y; a6 += e1.z; a7 += e1.w;
    }
    float4* o = (float4*)(esum + (size_t)gid * ED_);
    o[0] = make_float4(a0, a1, a2, a3);
    o[1] = make_float4(a4, a5, a6, a7);
}

// ---------------------------------------------------------------------------
// Kernel 2: persistent sequential scan. 16 blocks x 16 batch rows x 128 thr.
// Weights live in LDS for all 512 steps; GRU matmuls via v_wmma_f32_16x16x4_f32.
// Weight LDS layout is K-pair interleaved so each WMMA B fragment is one
// contiguous 8-byte LDS load straight into an even-aligned VGPR pair.
// ---------------------------------------------------------------------------
__global__ __launch_bounds__(SCAN_THREADS)
void pk_scan_kernel(const float* __restrict__ cont, const int* __restrict__ cat,
                    const float* __restrict__ lb,   const int* __restrict__ mask,
                    const float* __restrict__ d_arr, const float* __restrict__ td,
                    const float* __restrict__ W_ih, const float* __restrict__ W_hh,
                    const float* __restrict__ b_ih, const float* __restrict__ b_hh,
                    const float* __restrict__ W_out, const float* __restrict__ b_out,
                    const float* __restrict__ esum, float* __restrict__ out) {
    // Fused weight matrix Wc(k,n), stored as Wc2[k>>1][n][k&1]:
    //   n in [0,128):   k<31 -> W_ih[n][k];  31<=k<95 -> W_hh[n][k-31]   (r,z gates fused)
    //   n in [128,192): k<31 -> W_ih[n][k];  else 0                      (inn: xi part only)
    //   n in [192,256): 31<=k<95 -> W_hh[n-64][k-31]; else 0             (hn: h part only)
    __shared__ __align__(16) float Wc2[KTOT / 2][NTOT][2];   // 96 KB
    __shared__ __align__(16) float Xl[ROWS][KTOT];  // 0..29 feat, 30 vfb, 31..94 h, 95 pad
    __shared__ __align__(16) float Gl[ROWS][NTOT];  // matmul output
    __shared__ float biasl[NTOT];
    __shared__ float Woutl[4][H_];
    __shared__ float boutl[4];
    __shared__ float etal[ROWS][4];

    const int tid  = threadIdx.x;
    const int lane = tid & 31;
    const int wave = tid >> 5;
    const int row0 = blockIdx.x * ROWS;

    // ---- Phase 0: resident weights/biases into LDS, zero the X buffer ----
    for (int idx = tid; idx < KTOT * NTOT; idx += SCAN_THREADS) {
        int k = idx / NTOT, n = idx % NTOT;
        float v = 0.f;
        if (n < 128) {
            if (k < 31)       v = W_ih[n * 31 + k];
            else if (k < 95)  v = W_hh[n * 64 + (k - 31)];
        } else if (n < 192) {
            if (k < 31)       v = W_ih[n * 31 + k];
        } else {
            if (k >= 31 && k < 95) v = W_hh[(n - 64) * 64 + (k - 31)];
        }
        Wc2[k >> 1][n][k & 1] = v;
    }
    for (int n = tid; n < NTOT; n += SCAN_THREADS) {
        float v;
        if (n < 128)      v = b_ih[n] + b_hh[n];
        else if (n < 192) v = b_ih[n];            // inn bias
        else              v = b_hh[n - 64];       // hn bias
        biasl[n] = v;
    }
    for (int idx = tid; idx < 4 * H_; idx += SCAN_THREADS)
        Woutl[idx / H_][idx % H_] = W_out[idx];
    if (tid < 4) boutl[tid] = b_out[tid];
    for (int idx = tid; idx < ROWS * KTOT; idx += SCAN_THREADS)
        (&Xl[0][0])[idx] = 0.f;                   // h=0, vfb=0, pad=0

    // ---- Per-row recurrent state in registers of lanes 0..15 ----
    float Areg = 0.f, Creg = 0.f, Dreg = 0.f, cclbase = 0.f;
    const int brow = row0 + (tid < ROWS ? tid : 0);
    if (tid < ROWS) {
        const float* c0 = cont + (size_t)brow * T_ * NC_;   // t = 0
        float age    = c0[2]  * 16.936469f + 58.239251f;
        float weight = c0[21] * 30.519849f + 87.5752f;
        const int* catp = cat + (size_t)brow * T_ * L_;     // t = 0
        int gcnt = 0;
        for (int l = 0; l < L_; ++l) gcnt += (catp[l] == 5) ? 1 : 0;
        float gender = (float)gcnt * 0.15f + 0.85f;
        cclbase = (140.0f - age) * weight * gender * (1.0f / 72.0f);
    }
    __syncthreads();

    for (int t = 0; t < T_; ++t) {
        // ---- Phase A: stage X = [emb_sum(8) | cont(22) | vfb] ----
        for (int idx = tid; idx < ROWS * 30; idx += SCAN_THREADS) {
            int r = idx / 30, c = idx % 30;
            size_t bt = (size_t)(row0 + r) * T_ + t;
            Xl[r][c] = (c < ED_) ? esum[bt * ED_ + c]
                                 : cont[bt * NC_ + (c - ED_)];
        }
        if (tid < ROWS) {
            size_t bt = (size_t)brow * T_ + t;
            float mp = 0.f, lp = 0.f;
            if (t > 0) { mp = (float)mask[bt - 1]; lp = lb[bt - 1]; }
            Xl[tid][30] = Areg * (1.0f - mp) + lp * mp;     // vfb feedback
        }
        __syncthreads();

        // ---- Phase B: G[16][256] = X[16][96] @ Wc[96][256] via WMMA f32 ----
        {
            const int m  = lane & 15;
            const int hi = lane >> 4;         // lanes 16..31 carry K+2 / M+8 halves
            const int n0 = wave * 64;         // 4 N-tiles of 16 per wave
            v8f acc0 = {}, acc1 = {}, acc2 = {}, acc3 = {};
            for (int k0 = 0; k0 < KTOT; k0 += 4) {
                const int kp = (k0 >> 1) + hi;       // K-pair index for this lane half
                // A 16x4 fragment: lanes 0-15 hold K{0,1}, lanes 16-31 K{2,3}
                v2f a = *(const v2f*)&Xl[m][k0 + 2 * hi];
                // B 4x16 fragments: contiguous (K,K+1) pair per lane -> ds_load_b64
                v2f b0 = *(const v2f*)&Wc2[kp][n0 +  0 + m][0];
                v2f b1 = *(const v2f*)&Wc2[kp][n0 + 16 + m][0];
                v2f b2 = *(const v2f*)&Wc2[kp][n0 + 32 + m][0];
                v2f b3 = *(const v2f*)&Wc2[kp][n0 + 48 + m][0];
                acc0 = __builtin_amdgcn_wmma_f32_16x16x4_f32(false, a, false, b0, (short)0, acc0, false, false);
                acc1 = __builtin_amdgcn_wmma_f32_16x16x4_f32(false, a, false, b1, (short)0, acc1, false, false);
                acc2 = __builtin_amdgcn_wmma_f32_16x16x4_f32(false, a, false, b2, (short)0, acc2, false, false);
                acc3 = __builtin_amdgcn_wmma_f32_16x16x4_f32(false, a, false, b3, (short)0, acc3, false, false);
            }
            // C/D layout: VGPR i -> M = i (lanes 0-15) / i+8 (lanes 16-31), N = lane&15
            #pragma unroll
            for (int i = 0; i < 8; ++i) {
                int r = i + 8 * hi;
                Gl[r][n0 +  0 + m] = acc0[i];
                Gl[r][n0 + 16 + m] = acc1[i];
                Gl[r][n0 + 32 + m] = acc2[i];
                Gl[r][n0 + 48 + m] = acc3[i];
            }
        }
        __syncthreads();

        // ---- Phase C: GRU elementwise, h update in place (X cols 31..94) ----
        {
            const int c  = tid & 63;
            const int rb = tid >> 6;          // 0 or 1; rows rb, rb+2, ..., rb+14
            #pragma unroll
            for (int i = 0; i < 8; ++i) {
                int r = rb + 2 * i;
                float rv = Gl[r][c] + biasl[c];
                rv = 1.0f / (1.0f + expf(-rv));
                float zv = Gl[r][64 + c] + biasl[64 + c];
                zv = 1.0f / (1.0f + expf(-zv));
                float innv = Gl[r][128 + c] + biasl[128 + c];
                float hnv  = Gl[r][192 + c] + biasl[192 + c];
                float nv = tanhf(innv + rv * hnv);
                float hold = Xl[r][31 + c];
                Xl[r][31 + c] = (1.0f - zv) * nv + zv * hold;
            }
        }
        __syncthreads();

        // ---- Phase D: eta = h @ W_out^T + b_out  (16 rows x 4 outputs) ----
        if (tid < 64) {
            int r = tid >> 2, j = tid & 3;
            float s = boutl[j];
            #pragma unroll 8
            for (int c = 0; c < H_; ++c) s += Xl[r][31 + c] * Woutl[j][c];
            etal[r][j] = s;
        }
        __syncthreads();

        // ---- Phase E: 2-compartment PK update; carries in registers ----
        if (tid < ROWS) {
            size_t bt = (size_t)brow * T_ + t;
            float creat = cont[bt * NC_ + 11] * 1.418099f + 1.314668f;
            float cc = cclbase / creat;
            float e1 = etal[tid][0], e2 = etal[tid][1];
            float e3 = etal[tid][2], e4 = etal[tid][3];
            float v1  = 33.1f * expf(e1);
            float k1  = 3.96f * cc * 0.01f * expf(e2);
            float rr  = 1000.0f / v1;
            float v2  = 48.3f * expf(e4);
            float Rv  = (-6.99f / 48.3f) * expf(e3);
            float k2  = Rv * v2;
            float q   = k1 - k2 - Rv * v1;
            float dlt = sqrtf(q * q + 4.0f * k1 * Rv * v1) / v1;
            float sv  = (-k1 + k2 + Rv * v1) / v1;
            float lam1 = (sv - dlt) * 0.5f;
            float lam2 = (sv + dlt) * 0.5f;
            float kv  = k2 / v1;
            float C1  =  rr * kv / lam1 / dlt;
            float C2  = -rr * kv / lam2 / dlt;
            float C3  = -(lam1 - Rv) / kv;
            float C4  = -(lam2 - Rv) / kv;
            float dd  = d_arr[bt];
            float tt  = td[bt] * 24.0f;
            float Cn  = ((Creg + C1) * expf(lam1 * dd) - C1) * expf(lam1 * (tt - dd));
            float Dn  = ((Dreg + C2) * expf(lam2 * dd) - C2) * expf(lam2 * (tt - dd));
            float An  = Cn * C3 + Dn * C4;
            Creg = Cn; Dreg = Dn; Areg = An;
            out[bt] = An;                    // out is (B,T) row-major = A_seq.T
        }
        __syncthreads();
    }
}

// ---------------------------------------------------------------------------
extern "C" void kernel_launch(void* const* d_in, const int* in_sizes, int n_in,
                              void* d_out, int out_size, void* d_ws, size_t ws_size,
                              hipStream_t stream) {
    const float* cont  = (const float*)d_in[0];
    const int*   cat   = (const int*)  d_in[1];
    const float* lb    = (const float*)d_in[2];
    const int*   mask  = (const int*)  d_in[3];
    const float* d_arr = (const float*)d_in[4];
    const float* td    = (const float*)d_in[5];
    const float* emb   = (const float*)d_in[6];
    const float* W_ih  = (const float*)d_in[7];
    const float* W_hh  = (const float*)d_in[8];
    const float* b_ih  = (const float*)d_in[9];
    const float* b_hh  = (const float*)d_in[10];
    const float* W_out = (const float*)d_in[11];
    const float* b_out = (const float*)d_in[12];
    float* out  = (float*)d_out;
    float* esum = (float*)d_ws;              // B*T*8 f32 = 4 MB scratch

    int nbt = B_ * T_;
    pk_esum_kernel<<<(nbt + 255) / 256, 256, 0, stream>>>(cat, emb, esum);
    pk_scan_kernel<<<B_ / ROWS, SCAN_THREADS, 0, stream>>>(
        cont, cat, lb, mask, d_arr, td,
        W_ih, W_hh, b_ih, b_hh, W_out, b_out, esum, out);
}